// diffgait_denoising_branch_71227737637192
// MI455X (gfx1250) — compile-verified
//
#include <hip/hip_runtime.h>
#include <hip/hip_bf16.h>
#include <stdint.h>

#define NSTEPS    1000
#define NELEM     45056      // 16 * 64 * 44
#define HALF_ELEM 22528
#define COND_OFF  6000       // float offset of cond[] inside d_ws
// ws float layout: [0,1000)=A  [1000,2000)=B  [2000,3000)=temb
//                  [3000,4000)=c1  [4000,5000)=key0(u32)  [5000,6000)=key1(u32)
//                  [6000, 6000+45056) = cond

typedef float v2f __attribute__((ext_vector_type(2)));
typedef float v8f __attribute__((ext_vector_type(8)));

__device__ __forceinline__ uint32_t rotl32(uint32_t v, int r) {
  return (v << r) | (v >> (32 - r));
}

// JAX Threefry-2x32, 20 rounds.
__device__ __forceinline__ void threefry2x32(uint32_t k0, uint32_t k1,
                                             uint32_t x0, uint32_t x1,
                                             uint32_t& o0, uint32_t& o1) {
  const uint32_t ks0 = k0, ks1 = k1, ks2 = k0 ^ k1 ^ 0x1BD11BDAu;
  x0 += ks0; x1 += ks1;
#define TF_ROUND(r) { x0 += x1; x1 = rotl32(x1, r); x1 ^= x0; }
  TF_ROUND(13) TF_ROUND(15) TF_ROUND(26) TF_ROUND(6)
  x0 += ks1; x1 += ks2 + 1u;
  TF_ROUND(17) TF_ROUND(29) TF_ROUND(16) TF_ROUND(24)
  x0 += ks2; x1 += ks0 + 2u;
  TF_ROUND(13) TF_ROUND(15) TF_ROUND(26) TF_ROUND(6)
  x0 += ks0; x1 += ks1 + 3u;
  TF_ROUND(17) TF_ROUND(29) TF_ROUND(16) TF_ROUND(24)
  x0 += ks1; x1 += ks2 + 4u;
  TF_ROUND(13) TF_ROUND(15) TF_ROUND(26) TF_ROUND(6)
  x0 += ks2; x1 += ks0 + 5u;
#undef TF_ROUND
  o0 = x0; o1 = x1;
}

// JAX uniform(key, lo=nextafter(-1,0), hi=1) bit manipulation.
__device__ __forceinline__ float bits_to_uniform(uint32_t b) {
  float f = __uint_as_float((b >> 9) | 0x3f800000u) - 1.0f;  // [0,1)
  const float LO = -0.99999994f;                             // nextafterf(-1,0)
  float u = f * 2.0f + LO;                                   // (hi-lo) rounds to 2.0f
  return fmaxf(LO, u);
}

// XLA ErfInv32 (Giles).
__device__ __forceinline__ float erfinv_f32(float x) {
  float w = -__logf((1.0f - x) * (1.0f + x));
  float p;
  if (w < 5.0f) {
    w -= 2.5f;
    p = 2.81022636e-08f;
    p = fmaf(p, w, 3.43273939e-07f);
    p = fmaf(p, w, -3.5233877e-06f);
    p = fmaf(p, w, -4.39150654e-06f);
    p = fmaf(p, w, 0.00021858087f);
    p = fmaf(p, w, -0.00125372503f);
    p = fmaf(p, w, -0.00417768164f);
    p = fmaf(p, w, 0.246640727f);
    p = fmaf(p, w, 1.50140941f);
  } else {
    w = __fsqrt_rn(w) - 3.0f;
    p = -0.000200214257f;
    p = fmaf(p, w, 0.000100950558f);
    p = fmaf(p, w, 0.00134934322f);
    p = fmaf(p, w, -0.00367342844f);
    p = fmaf(p, w, 0.00573950773f);
    p = fmaf(p, w, -0.0076224613f);
    p = fmaf(p, w, 0.00943887047f);
    p = fmaf(p, w, 1.00167406f);
    p = fmaf(p, w, 2.83297682f);
  }
  return p * x;
}

__device__ __forceinline__ float fast_tanh(float x) {
#if __has_builtin(__builtin_amdgcn_tanhf)
  return __builtin_amdgcn_tanhf(x);   // gfx1250 v_tanh_f32
#else
  return tanhf(x);
#endif
}

// ---------------- Kernel 1: DDIM schedule + per-step folded keys ----------------
__global__ void ddim_schedule_kernel(const float* __restrict__ t_scale,
                                     float* __restrict__ wsf) {
  int j = blockIdx.x * blockDim.x + threadIdx.x;   // scan step 0..999
  if (j >= NSTEPS) return;
  int t = 999 - j;                                 // diffusion index (reversed scan)
  const double b0 = 1e-4, db = (0.02 - 1e-4) / 999.0;
  double p = 1.0, alph_t = 1.0;                    // alph[t], alph[t+1]
  for (int i = 0; i <= t; ++i) {
    if (i == t) alph_t = p;
    p *= 1.0 - (b0 + db * (double)i);
  }
  float at  = (float)p;        // alph[t+1]
  float atn = (float)alph_t;   // alph[t]
  float om_at = 1.0f - at;
  float c1 = 0.1f * sqrtf((1.0f - at / atn) * (1.0f - atn) / om_at);
  float c2 = sqrtf(fmaxf(1.0f - atn - c1 * c1, 0.0f));
  float inv_s = 1.0f / sqrtf(om_at);
  float A = sqrtf(atn) - c2 * sqrtf(at) * inv_s;
  float B = c2 * inv_s;
  float temb = t_scale[0] * ((float)t / 1000.0f);

  uint32_t fk0, fk1;
  threefry2x32(0u, 42u, 0u, (uint32_t)t, fk0, fk1);   // fold_in(key(42), t)

  wsf[j]        = A;
  wsf[1000 + j] = B;
  wsf[2000 + j] = temb;
  wsf[3000 + j] = c1;
  uint32_t* wsu = (uint32_t*)wsf;
  wsu[4000 + j] = fk0;
  wsu[5000 + j] = fk1;
}

// ---------------- Kernel 2: cond = einsum('nchw,c') via WMMA 16x16x4 f32 ----------------
__global__ void cond_wmma_kernel(const float* __restrict__ skes,
                                 const float* __restrict__ wsk,
                                 float* __restrict__ wsf) {
  int gid  = blockIdx.x * blockDim.x + threadIdx.x;
  int wave = gid >> 5;
  int lane = threadIdx.x & 31;
  int e_base = wave * 16;                 // this wave produces cond[e_base .. e_base+15]
  int m   = lane & 15;
  int row = e_base + m;                   // element index in (n*s, h, w) flattening
  int hw  = row % 2816;                   // h*44 + w
  int ns  = row / 2816;
  int n   = ns >> 3, s = ns & 7;
  int base = (n * 24 + s) * 2816 + hw;    // skes flat index at c=0; c-stride = 22528
  bool lo = lane < 16;

  // A (16x4, f32): lanes 0-15 hold K=0 (v0) / K=1 (v1); lanes 16-31 hold K=2 / K=3(=0 pad)
  float a0 = skes[base + (lo ? 0 : 45056)];
  float a1 = lo ? skes[base + 22528] : 0.0f;
  // B (4x16): every column equals w_skes (k=3 padded 0) -> D[m,n] = cond[m] for all n
  float b0 = lo ? wsk[0] : wsk[2];
  float b1 = lo ? wsk[1] : 0.0f;

  float* condp = wsf + COND_OFF;
#if __has_builtin(__builtin_amdgcn_wmma_f32_16x16x4_f32)
  v2f Av = {a0, a1};
  v2f Bv = {b0, b1};
  v8f Cv = {0.f, 0.f, 0.f, 0.f, 0.f, 0.f, 0.f, 0.f};
  v8f Dv = __builtin_amdgcn_wmma_f32_16x16x4_f32(false, Av, false, Bv,
                                                 (short)0, Cv, false, false);
  // C/D layout: lanes 0-15 vgpr r -> M=r ; lanes 16-31 vgpr r -> M=r+8 (N replicated)
  if (lane == 0) {
#pragma unroll
    for (int r = 0; r < 8; ++r) condp[e_base + r] = Dv[r];
  } else if (lane == 16) {
#pragma unroll
    for (int r = 0; r < 8; ++r) condp[e_base + 8 + r] = Dv[r];
  }
#else
  if (lo) condp[row] = skes[base] * wsk[0] + skes[base + 22528] * wsk[1]
                     + skes[base + 45056] * wsk[2];
  (void)a0; (void)a1; (void)b0; (void)b1;
#endif
}

// ---------------- Kernel 3: full-occupancy noise: d_out[j,e] = sqrt2*c1_j*erfinv(u) ----------------
__global__ void noise_kernel(const float* __restrict__ wsf,
                             float* __restrict__ out) {
  int e = blockIdx.x * blockDim.x + threadIdx.x;   // 0..22527 (low half)
  int j = blockIdx.y;                              // scan step
  const uint32_t* wsu = (const uint32_t*)wsf;
  uint32_t k0 = wsu[4000 + j], k1 = wsu[5000 + j];
  float scale = wsf[3000 + j] * 1.41421356237309504880f;  // sqrt(2)*c1

  uint32_t r0, r1;
  threefry2x32(k0, k1, (uint32_t)e, (uint32_t)(e + HALF_ELEM), r0, r1);
  float n0 = scale * erfinv_f32(bits_to_uniform(r0));
  float n1 = scale * erfinv_f32(bits_to_uniform(r1));

  size_t o = (size_t)j * NELEM + (size_t)e;
  out[o] = n0;
  out[o + HALF_ELEM] = n1;
}

// ---------------- Kernel 4: sequential 1000-step chain; overwrite noise with sil_0 ----------------
__global__ void chain_kernel(const float* __restrict__ silT,
                             const float* __restrict__ w_sil,
                             const float* __restrict__ wsf,
                             float* __restrict__ out) {
  // sC: [0,1000)=A  [1000,2000)=B  [2000,3000)=temb  (same order as ws)
  __shared__ float sC[3 * NSTEPS];
  int tid = threadIdx.x;
#if defined(__gfx1250__)
  // CDNA5 async copy: each lane issues GLOBAL_LOAD_ASYNC_TO_LDS_B32 (GVS mode:
  // saddr = ws base, vaddr = byte offset; LDS dest = low 32 bits of generic
  // shared pointer == LDS byte offset per the aperture address rules).
  {
    uint64_t gbase = (uint64_t)(uintptr_t)wsf;
    for (int i = tid; i < 3 * NSTEPS; i += 256) {
      uint32_t lds_addr = (uint32_t)(uintptr_t)(&sC[i]);
      uint32_t goff     = (uint32_t)(i * 4);
      asm volatile("global_load_async_to_lds_b32 %0, %1, %2"
                   :: "v"(lds_addr), "v"(goff), "s"(gbase)
                   : "memory");
    }
    asm volatile("s_wait_asynccnt 0x0" ::: "memory");
  }
#else
  for (int i = tid; i < 3 * NSTEPS; i += 256) sC[i] = wsf[i];
#endif
  __syncthreads();

  int e = blockIdx.x * 256 + tid;                  // 0..22527
  float wsil = w_sil[0];
  float ca = wsf[COND_OFF + e];
  float cb = wsf[COND_OFF + e + HALF_ELEM];
  float Ta = silT[e];
  float Tb = silT[e + HALF_ELEM];
  float xa = Ta, xb = Tb;

  float* po = out + e;
#pragma unroll 4
  for (int j = 0; j < NSTEPS; ++j) {
    float Aj = sC[j], Bj = sC[1000 + j], Tj = sC[2000 + j];
    float s0a = fast_tanh(fmaf(wsil, xa, ca) + Tj);
    float s0b = fast_tanh(fmaf(wsil, xb, cb) + Tj);
    // noise (c1*eps) lives at the preds slot; read it, then overwrite with sil_0.
    float na = __builtin_nontemporal_load(po);
    float nb = __builtin_nontemporal_load(po + HALF_ELEM);
    // fma(0, noise, s0) == s0 but makes the store depend on the load -> read-before-write.
    __builtin_nontemporal_store(fmaf(0.0f, na, s0a), po);
    __builtin_nontemporal_store(fmaf(0.0f, nb, s0b), po + HALF_ELEM);
    xa = fmaf(Aj, s0a, fmaf(Bj, Ta, na));
    xb = fmaf(Aj, s0b, fmaf(Bj, Tb, nb));
    po += NELEM;
  }
}

extern "C" void kernel_launch(void* const* d_in, const int* in_sizes, int n_in,
                              void* d_out, int out_size, void* d_ws, size_t ws_size,
                              hipStream_t stream) {
  const float* skes    = (const float*)d_in[0];   // (2,3,8,64,44)
  const float* silT    = (const float*)d_in[1];   // (16,1,64,44)
  const float* wsk     = (const float*)d_in[2];   // (3,)
  const float* w_sil   = (const float*)d_in[3];   // scalar
  const float* t_scale = (const float*)d_in[4];   // scalar
  float* out = (float*)d_out;                     // (1000,16,1,64,44)
  float* wsf = (float*)d_ws;                      // ~204 KB used

  hipLaunchKernelGGL(ddim_schedule_kernel, dim3(4), dim3(256), 0, stream,
                     t_scale, wsf);
  hipLaunchKernelGGL(cond_wmma_kernel, dim3(352), dim3(256), 0, stream,
                     skes, wsk, wsf);
  hipLaunchKernelGGL(noise_kernel, dim3(88, 1000), dim3(256), 0, stream,
                     wsf, out);
  hipLaunchKernelGGL(chain_kernel, dim3(88), dim3(256), 0, stream,
                     silT, w_sil, wsf, out);
}